// GRNCausalRegularizer_7980049236665
// MI455X (gfx1250) — compile-verified
//
#include <hip/hip_runtime.h>

typedef __attribute__((ext_vector_type(2))) float v2f;
typedef __attribute__((ext_vector_type(8))) float v8f;

#define NB 8192   // batch rows
#define NG 4096   // genes (columns / adjacency dim)

// ---------------------------------------------------------------------------
// ws layout (floats): [0, NG)  -> v_mag accumulators
//                     [NG]     -> total (weighted edge sum)
//                     [NG+1]   -> n_edges
// ---------------------------------------------------------------------------

__global__ void grn_init_ws(float* __restrict__ ws) {
    int i = blockIdx.x * blockDim.x + threadIdx.x;
    if (i < NG + 2) ws[i] = 0.0f;
}

// Column abs-sum of velocity via V_WMMA_F32_16X16X4_F32:
//   D = ones(16x4) @ |vel tile|(4x16) + C   -> every row of D = column sums.
// Each wave owns a 16-column strip and a contiguous row chunk.
// blockDim = 256 (8 waves); wave w covers columns (blockIdx.x*8 + w)*16.
__global__ __launch_bounds__(256) void grn_vmag_wmma(const float* __restrict__ vel,
                                                     float* __restrict__ vmag,
                                                     int rowsPerWave) {
    const int lane  = threadIdx.x & 31;
    const int wave  = threadIdx.x >> 5;
    const int col0  = (blockIdx.x * 8 + wave) * 16;
    const int row0  = blockIdx.y * rowsPerWave;

    const int n     = lane & 15;   // column within strip
    const int khalf = lane >> 4;   // 0: K={0,1}, 1: K={2,3}  (f32 A/B VGPR layout)

    // A = all ones (16x4 f32): both A VGPRs are 1.0 in every lane.
    v2f a; a[0] = 1.0f; a[1] = 1.0f;

    // Two independent accumulators to break the WMMA->WMMA RAW chain.
    v8f c0 = {};
    v8f c1 = {};

    // B reg0 <- row (khalf*2), B reg1 <- row (khalf*2 + 1) of the 4x16 tile.
    const float* p = vel + ((size_t)row0 + (size_t)(khalf << 1)) * NG + col0 + n;
    const size_t rowStride = (size_t)NG;

    for (int r = 0; r < rowsPerWave; r += 8) {
        // tile 0 (rows r .. r+3)
        v2f b;
        b[0] = fabsf(p[0]);
        b[1] = fabsf(p[rowStride]);
        c0 = __builtin_amdgcn_wmma_f32_16x16x4_f32(false, a, false, b,
                                                   (short)0, c0, false, false);
        // tile 1 (rows r+4 .. r+7)
        const float* q = p + 4 * rowStride;
        v2f b2;
        b2[0] = fabsf(q[0]);
        b2[1] = fabsf(q[rowStride]);
        c1 = __builtin_amdgcn_wmma_f32_16x16x4_f32(false, a, false, b2,
                                                   (short)0, c1, false, false);
        p += 8 * rowStride;
        __builtin_prefetch(p, 0, 1);  // global_prefetch_b8 (speculative, safe)
    }

    // D layout: VGPR0, lanes 0..15 hold D[0][n] == column sum for col0+n.
    float colsum = c0[0] + c1[0];
    if (lane < 16) {
        atomicAdd(&vmag[col0 + lane], colsum * (1.0f / (float)NB));
    }
}

// Pairwise edge pass: stream grn_adj once (float4), v_mag cached in LDS.
__global__ __launch_bounds__(256) void grn_pair_sum(const float* __restrict__ adj,
                                                    const float* __restrict__ vmag,
                                                    float* __restrict__ acc) {
    __shared__ float s_vmag[NG];
    __shared__ float s_red[2][8];

    for (int i = threadIdx.x; i < NG; i += blockDim.x)
        s_vmag[i] = vmag[i];
    __syncthreads();

    float tsum = 0.0f;
    float tcnt = 0.0f;

    const float4* adj4  = (const float4*)adj;
    const size_t  total4 = ((size_t)NG * NG) >> 2;
    const size_t  stride = (size_t)gridDim.x * blockDim.x;

    for (size_t q = (size_t)blockIdx.x * blockDim.x + threadIdx.x;
         q < total4; q += stride) {
        float4 a4 = adj4[q];
        __builtin_prefetch(adj4 + q + stride, 0, 1);
        const size_t idx = q << 2;
        const int i  = (int)(idx >> 12);     // row  (TF)      : idx / 4096
        const int j0 = (int)(idx & 4095);    // col0 (target)  : 4 consecutive
        const float mi = s_vmag[i];

        float av[4] = {a4.x, a4.y, a4.z, a4.w};
        #pragma unroll
        for (int e = 0; e < 4; ++e) {
            const float a = av[e];
            if (a != 0.0f) {
                const float mj = s_vmag[j0 + e];
                tsum += (a > 0.0f) ? fabsf(mj - mi) : fabsf(mj + mi);
                tcnt += 1.0f;
            }
        }
    }

    // wave32 reduction
    #pragma unroll
    for (int off = 16; off > 0; off >>= 1) {
        tsum += __shfl_down(tsum, off, 32);
        tcnt += __shfl_down(tcnt, off, 32);
    }
    const int lane = threadIdx.x & 31;
    const int wave = threadIdx.x >> 5;
    if (lane == 0) { s_red[0][wave] = tsum; s_red[1][wave] = tcnt; }
    __syncthreads();
    if (wave == 0) {
        float bs = (lane < 8) ? s_red[0][lane] : 0.0f;
        float bc = (lane < 8) ? s_red[1][lane] : 0.0f;
        #pragma unroll
        for (int off = 4; off > 0; off >>= 1) {
            bs += __shfl_down(bs, off, 32);
            bc += __shfl_down(bc, off, 32);
        }
        if (lane == 0) {
            atomicAdd(&acc[0], bs);
            atomicAdd(&acc[1], bc);
        }
    }
}

__global__ void grn_finalize(const float* __restrict__ acc, float* __restrict__ out) {
    const float total = acc[0];
    const float n     = acc[1];
    out[0] = (n > 0.0f) ? (total / fmaxf(n, 1.0f)) : 0.0f;
}

extern "C" void kernel_launch(void* const* d_in, const int* in_sizes, int n_in,
                              void* d_out, int out_size, void* d_ws, size_t ws_size,
                              hipStream_t stream) {
    (void)in_sizes; (void)n_in; (void)out_size; (void)ws_size;
    const float* vel = (const float*)d_in[0];   // velocity (B, G) f32
    // d_in[1] = perturbation_idx — unused by the reference output
    const float* adj = (const float*)d_in[2];   // grn_adj (G, G) f32 in {-1,0,1}
    float* ws  = (float*)d_ws;
    float* out = (float*)d_out;

    // 1) zero v_mag + accumulators (graph-safe, deterministic per call)
    grn_init_ws<<<(NG + 2 + 255) / 256, 256, 0, stream>>>(ws);

    // 2) v_mag via f32 WMMA column reduction
    //    grid.x: 4096 cols / (8 waves * 16 cols) = 32; grid.y: 8192 / 1024 = 8
    dim3 g1(NG / (8 * 16), NB / 1024);
    grn_vmag_wmma<<<g1, 256, 0, stream>>>(vel, ws, 1024);

    // 3) stream adjacency, accumulate weighted sum + edge count
    grn_pair_sum<<<1024, 256, 0, stream>>>(adj, ws, ws + NG);

    // 4) final scalar
    grn_finalize<<<1, 1, 0, stream>>>(ws + NG, out);
}